// AttSampler_51058571215050
// MI455X (gfx1250) — compile-verified
//
#include <hip/hip_runtime.h>
#include <hip/hip_bf16.h>
#include <stdint.h>

// Problem constants (from reference: SCALE=1.0, DENSE=4, ITERS=5)
#define AW   512          // attention length A == H == W
#define OUTS 512          // out_size = H * SCALE
#define NCH  3            // channels
#define NITER 5

// ---------------------------------------------------------------------------
// gfx1250 async global->LDS copy helpers (CDNA5 async/tensor data path).
// Each lane supplies its own LDS byte address (VGPR) and 64-bit global addr.
// Tracked by ASYNCcnt; drained with s_wait_asynccnt.
// ---------------------------------------------------------------------------
__device__ __forceinline__ void async_load_b128_to_lds(uint32_t lds_byte_off,
                                                       const float* gaddr) {
    asm volatile("global_load_async_to_lds_b128 %0, %1, off"
                 :
                 : "v"(lds_byte_off), "v"(gaddr)
                 : "memory");
}

__device__ __forceinline__ void wait_asynccnt0() {
#if __has_builtin(__builtin_amdgcn_s_wait_asynccnt)
    __builtin_amdgcn_s_wait_asynccnt(0);
#else
    asm volatile("s_wait_asynccnt 0x0" ::: "memory");
#endif
}

// ---------------------------------------------------------------------------
// Block-wide reduction over 512 threads through a shared 512-float buffer.
// Leading barrier protects previous use of buf.
// ---------------------------------------------------------------------------
template <bool IS_MAX>
__device__ __forceinline__ float block_reduce512(float v, float* buf, int tid) {
    __syncthreads();
    buf[tid] = v;
    __syncthreads();
#pragma unroll
    for (int s = 256; s > 0; s >>= 1) {
        if (tid < s) {
            float a = buf[tid], b = buf[tid + s];
            buf[tid] = IS_MAX ? fmaxf(a, b) : (a + b);
        }
        __syncthreads();
    }
    return buf[0];
}

// Inclusive scan of 512 values (Hillis-Steele) into buf, then per-thread
// inverse-CDF point: searchsorted(c, (tid+0.5)*c[511]/512, left), interp.
__device__ __forceinline__ float inverse_cdf_point(float v, float* buf, int tid) {
    __syncthreads();           // protect previous use of buf
    buf[tid] = v;
#pragma unroll
    for (int off = 1; off < AW; off <<= 1) {
        __syncthreads();
        float add = (tid >= off) ? buf[tid - off] : 0.0f;
        __syncthreads();
        buf[tid] += add;
    }
    __syncthreads();
    float total = buf[AW - 1];
    float step  = total * (1.0f / (float)OUTS);
    float tt    = ((float)tid + 0.5f) * step;
    // searchsorted side='left': first j in [0,512] with c[j] >= tt
    int l = 0, r = AW;
    while (l < r) {
        int m = (l + r) >> 1;          // m <= 511 always
        if (buf[m] < tt) l = m + 1; else r = m;
    }
    int j = (l < AW - 1) ? l : (AW - 1);   // clip to [0, 511]
    float c_prev = (j > 0) ? buf[j - 1] : 0.0f;
    float dens   = buf[j] - c_prev;
    float p      = (float)j + (tt - c_prev) / fmaxf(dens, 1e-6f);
    return p * (2.0f / (float)AW) - 1.0f;
}

// ---------------------------------------------------------------------------
// Kernel 1: iterative clip + inverse CDF. One block per batch, 512 threads,
// thread t owns attx[n,t] / atty[n,t].
// ---------------------------------------------------------------------------
__global__ __launch_bounds__(512)
void AttSampler_clip_cdf_kernel(const float* __restrict__ attx,
                                const float* __restrict__ atty,
                                float* __restrict__ pxbuf,
                                float* __restrict__ pybuf) {
    __shared__ float buf[AW];
    const int n   = blockIdx.x;
    const int tid = threadIdx.x;

    float ax = attx[n * AW + tid];
    float ay = atty[n * AW + tid];

    // normalize: a = a / sum(a) * out_size
    float sx = block_reduce512<false>(ax, buf, tid);
    ax = ax / sx * (float)OUTS;
    float sy = block_reduce512<false>(ay, buf, tid);
    ay = ay / sy * (float)OUTS;

    const float thr0 = (float)(4 * OUTS) / (float)AW;   // DENSE*out/A = 4.0
#pragma unroll
    for (int i = 0; i < NITER; ++i) {
        float mx = block_reduce512<true>(ax, buf, tid);
        float my = block_reduce512<true>(ay, buf, tid);
        float t  = fminf(mx, my);
        if (i == 0) t = fminf(t, thr0);
        ax = fminf(ax, t);
        ay = fminf(ay, t);
        float s1 = block_reduce512<false>(ax, buf, tid);
        ax += ((float)OUTS - s1) * (1.0f / (float)AW);
        float s2 = block_reduce512<false>(ay, buf, tid);
        ay += ((float)OUTS - s2) * (1.0f / (float)AW);
    }

    pxbuf[n * OUTS + tid] = inverse_cdf_point(ax, buf, tid);
    pybuf[n * OUTS + tid] = inverse_cdf_point(ay, buf, tid);
}

// ---------------------------------------------------------------------------
// Kernel 2: bilinear sampling. One block per (n, oy); stage the two source
// rows of all 3 channels in LDS via CDNA5 async b128 global->LDS loads, then
// each thread (== ox) gathers its 4 neighbors per channel from LDS.
// ---------------------------------------------------------------------------
__global__ __launch_bounds__(512)
void AttSampler_sample_kernel(const float* __restrict__ data,
                              const float* __restrict__ pxbuf,
                              const float* __restrict__ pybuf,
                              float* __restrict__ sampled,
                              float* __restrict__ grid_out) {
    __shared__ float rows[6 * AW];            // [c*2 + (top/bot)][x], 12 KB
    const int bid = blockIdx.x;
    const int n   = bid >> 9;                 // /512
    const int oy  = bid & (OUTS - 1);
    const int tid = threadIdx.x;              // == ox

    const float pyv = pybuf[n * OUTS + oy];
    const float iy  = (pyv + 1.0f) * 0.5f * (float)(AW - 1);
    const float y0f = floorf(iy);
    const float wy  = iy - y0f;
    int y0i = (int)y0f; y0i = y0i < 0 ? 0 : (y0i > AW - 1 ? AW - 1 : y0i);
    const int y1i = (y0i + 1 > AW - 1) ? (AW - 1) : (y0i + 1);

    // async-stage 6 rows (3 channels x {y0,y1}) into LDS as 768 x 16B chunks.
    // chunk k: row r = k>>7 (128 chunks/row), x-offset = (k&127)*4 floats.
    const uint32_t lds_base = (uint32_t)(uintptr_t)(void*)rows;  // low 32 bits = LDS offset
#pragma unroll
    for (int part = 0; part < 2; ++part) {
        const int k = part * 512 + tid;
        if (k < 6 * 128) {
            const int r  = k >> 7;
            const int c  = r >> 1;
            const int yy = (r & 1) ? y1i : y0i;
            const float* g =
                data + (((size_t)((n * NCH + c) * AW + yy)) << 9) + ((k & 127) << 2);
            const uint32_t loff = lds_base + ((uint32_t)k << 4);
            async_load_b128_to_lds(loff, g);
        }
    }
    wait_asynccnt0();
    __syncthreads();

    const float pxv = pxbuf[n * OUTS + tid];
    const float ix  = (pxv + 1.0f) * 0.5f * (float)(AW - 1);
    const float x0f = floorf(ix);
    const float wx  = ix - x0f;
    int x0i = (int)x0f; x0i = x0i < 0 ? 0 : (x0i > AW - 1 ? AW - 1 : x0i);
    const int x1i = (x0i + 1 > AW - 1) ? (AW - 1) : (x0i + 1);

    // ((n*3 + c)*512 + oy)*512 + ox ; c stride = 512*512 = 1<<18
    const size_t obase = (((size_t)n * NCH) << 18) + ((size_t)oy << 9) + (size_t)tid;
#pragma unroll
    for (int c = 0; c < NCH; ++c) {
        const float* top_row = &rows[(c * 2 + 0) << 9];
        const float* bot_row = &rows[(c * 2 + 1) << 9];
        const float v00 = top_row[x0i], v01 = top_row[x1i];
        const float v10 = bot_row[x0i], v11 = bot_row[x1i];
        const float top = v00 * (1.0f - wx) + v01 * wx;
        const float bot = v10 * (1.0f - wx) + v11 * wx;
        const float val = top * (1.0f - wy) + bot * wy;
        sampled[obase + ((size_t)c << 18)] = val;
    }

    // grid[n, oy, ox, 0:2] = (px, py)
    float2 g2; g2.x = pxv; g2.y = pyv;
    reinterpret_cast<float2*>(grid_out)[(((size_t)n << 9) + oy) * OUTS + tid] = g2;
}

// ---------------------------------------------------------------------------
// Launch wrapper
// ---------------------------------------------------------------------------
extern "C" void kernel_launch(void* const* d_in, const int* in_sizes, int n_in,
                              void* d_out, int out_size, void* d_ws, size_t ws_size,
                              hipStream_t stream) {
    const float* data = (const float*)d_in[0];   // (N, 3, 512, 512)
    const float* attx = (const float*)d_in[1];   // (N, 512, 1)
    const float* atty = (const float*)d_in[2];   // (N, 512, 1)

    const int N = in_sizes[1] / AW;              // 32

    float* pxbuf = (float*)d_ws;                 // N*512 floats
    float* pybuf = pxbuf + (size_t)N * OUTS;     // N*512 floats

    float* out     = (float*)d_out;
    float* sampled = out;                                          // N*3*512*512
    float* grid    = out + (size_t)N * NCH * OUTS * OUTS;          // N*512*512*2

    AttSampler_clip_cdf_kernel<<<N, 512, 0, stream>>>(attx, atty, pxbuf, pybuf);
    AttSampler_sample_kernel<<<N * OUTS, 512, 0, stream>>>(data, pxbuf, pybuf,
                                                           sampled, grid);
}